// DKWinners_80109730005713
// MI455X (gfx1250) — compile-verified
//
#include <hip/hip_runtime.h>
#include <stdint.h>

// ---------------------------------------------------------------------------
// DKWinners: per-16-element-segment argmax one-hot mask * x.
// B=1024, N=65536 (4096 segments x 16). Pure streaming: 512MB traffic,
// ~22us floor at 23.3 TB/s. CDNA5 path: async global->LDS double-buffered
// pipeline per wave (ASYNCcnt), GVS addressing with immediate offsets,
// 4 lanes/segment, value-only shuffle max-reduce + ballot winner pick,
// non-temporal b128 loads and stores.
// ---------------------------------------------------------------------------

typedef float v4f __attribute__((ext_vector_type(4)));

#define THREADS          256      // 8 waves (wave32)
#define J_PER_LANE       8        // float4s per lane per stage
#define F4_PER_WAVE      (32 * J_PER_LANE)        // 256 float4s / wave / stage
#define F4_PER_STAGE     (8 * F4_PER_WAVE)        // 2048 float4s = 32KB / block / stage
#define STAGES_PER_BLOCK 4

// GVS-mode async copy, 16B per lane. The instruction's IOFFSET is added to
// BOTH the LDS address (vdst VGPR) and the global address (saddr+voffset),
// so one immediate strides both sides identically.
#define ASYNC_LD16(ldsb, goff, gbase, IMM)                                     \
    asm volatile("global_load_async_to_lds_b128 %0, %1, %2 offset:" #IMM       \
                 " th:TH_LOAD_NT"                                              \
                 :                                                             \
                 : "v"(ldsb), "v"(goff), "s"(gbase)                            \
                 : "memory")

#define ISSUE_STAGE(ldsb, goff, gbase)                                         \
    do {                                                                       \
        ASYNC_LD16(ldsb, goff, gbase, 0);                                      \
        ASYNC_LD16(ldsb, goff, gbase, 512);                                    \
        ASYNC_LD16(ldsb, goff, gbase, 1024);                                   \
        ASYNC_LD16(ldsb, goff, gbase, 1536);                                   \
        ASYNC_LD16(ldsb, goff, gbase, 2048);                                   \
        ASYNC_LD16(ldsb, goff, gbase, 2560);                                   \
        ASYNC_LD16(ldsb, goff, gbase, 3072);                                   \
        ASYNC_LD16(ldsb, goff, gbase, 3584);                                   \
    } while (0)

__global__ __launch_bounds__(THREADS)
void DKWinners_80109730005713_kernel(const float* __restrict__ in,
                                     float* __restrict__ outp)
{
    __shared__ v4f lbuf[2][F4_PER_STAGE];   // 64 KB, double buffered

    const int lane  = threadIdx.x & 31;
    const int wave  = threadIdx.x >> 5;
    const int wbase = wave * F4_PER_WAVE;   // this wave's float4 region in a stage
    const unsigned q = (unsigned)(lane & 3);   // my quarter of the segment

    const v4f* __restrict__ in4  = (const v4f*)in;
    v4f* __restrict__       out4 = (v4f*)outp;

    const size_t blockBase =
        (size_t)blockIdx.x * (size_t)STAGES_PER_BLOCK * (size_t)F4_PER_STAGE;

    // Loop-invariant address components.
    const uint32_t goff  = (uint32_t)((wbase + lane) * 16);            // byte off in stage
    const uint32_t ldsb0 = (uint32_t)(uintptr_t)&lbuf[0][wbase + lane];
    const uint32_t ldsb1 = (uint32_t)(uintptr_t)&lbuf[1][wbase + lane];

    // Prime: stage 0 -> buffer 0.
    {
        const uint64_t gbase = (uint64_t)(uintptr_t)(in4 + blockBase);
        ISSUE_STAGE(ldsb0, goff, gbase);
    }

    for (int s = 0; s < STAGES_PER_BLOCK; ++s) {
        const int cur = s & 1;

        if (s + 1 < STAGES_PER_BLOCK) {
            // Fill the other buffer, then wait until only those 8 new loads
            // remain outstanding (async loads complete in order per wave)
            // => current buffer is resident.
            const uint64_t gbase = (uint64_t)(uintptr_t)
                (in4 + blockBase + (size_t)(s + 1) * F4_PER_STAGE);
            const uint32_t ldsn = (cur == 0) ? ldsb1 : ldsb0;
            ISSUE_STAGE(ldsn, goff, gbase);
            asm volatile("s_wait_asynccnt 8" ::: "memory");
        } else {
            asm volatile("s_wait_asynccnt 0" ::: "memory");
        }

        const size_t sb = blockBase + (size_t)s * F4_PER_STAGE;
        #pragma unroll
        for (int j = 0; j < J_PER_LANE; ++j) {
            const int li = wbase + j * 32 + lane;

            // Lane-linear LDS read: conflict-free ds_load_b128.
            v4f v = lbuf[cur][li];

            // Segment max: local fmax chain + 2 value-only shuffle rounds
            // across the aligned 4-lane group.
            float mv = fmaxf(fmaxf(v.x, v.y), fmaxf(v.z, v.w));
            mv = fmaxf(mv, __shfl_xor(mv, 1, 32));
            mv = fmaxf(mv, __shfl_xor(mv, 2, 32));

            // Winner quarter = lowest lane in group whose quarter holds the max
            // (fmax returns an input bit-exactly, so == is sound; lowest lane +
            // lowest local slot reproduces jnp.argmax first-occurrence ties).
            const unsigned long long bal =
                __ballot(v.x == mv || v.y == mv || v.z == mv || v.w == mv);
            const unsigned grp = ((unsigned)(bal >> (lane & ~3))) & 0xFu;
            const unsigned wq  = (unsigned)__builtin_ctz(grp);

            v4f o = (v4f)0.0f;
            if (wq == q) {
                // Winner lives in my quarter: keep only the first local match.
                if (v.x == mv)      o.x = v.x;
                else if (v.y == mv) o.y = v.y;
                else if (v.z == mv) o.z = v.z;
                else                o.w = v.w;
            }

            // Streaming output: non-temporal b128 store.
            __builtin_nontemporal_store(o, out4 + sb + li);
        }
    }
}

extern "C" void kernel_launch(void* const* d_in, const int* in_sizes, int n_in,
                              void* d_out, int out_size, void* d_ws, size_t ws_size,
                              hipStream_t stream) {
    (void)n_in; (void)out_size; (void)d_ws; (void)ws_size;

    const float* x   = (const float*)d_in[0];
    float*       out = (float*)d_out;

    const long long n           = (long long)in_sizes[0];   // 1024*65536
    const long long nF4         = n >> 2;                    // float4 count
    const long long totalStages = nF4 / F4_PER_STAGE;        // 8192
    const int       blocks      = (int)(totalStages / STAGES_PER_BLOCK); // 2048

    DKWinners_80109730005713_kernel<<<blocks, THREADS, 0, stream>>>(x, out);
}